// ObjectEncoder_8323646619965
// MI455X (gfx1250) — compile-verified
//
#include <hip/hip_runtime.h>
#include <hip/hip_bf16.h>

typedef __attribute__((ext_vector_type(2))) float v2f;
typedef __attribute__((ext_vector_type(8))) float v8f;

#define BATCH 16
#define NPTS  8192

// D = A(16x4,f32) * B(4x16,f32) + C(16x16,f32)  -- pure f32 matrix pipe
__device__ __forceinline__ v8f wmma4(v2f a, v2f b, v8f c) {
  return __builtin_amdgcn_wmma_f32_16x16x4_f32(
      /*neg_a=*/false, a, /*neg_b=*/false, b,
      /*c_mod=*/(short)0, c, /*reuse_a=*/false, /*reuse_b=*/false);
}

// ---------------------------------------------------------------------------
// Farthest point sampling: 1 workgroup per batch, 256 threads, PPT pts/thread.
// Exact jnp.argmax semantics: max value, ties -> lowest index.
// ---------------------------------------------------------------------------
template<int PPT>
__global__ void fps_kernel(const float* __restrict__ xyz, int S,
                           float* __restrict__ ctrs) {
  const int N = PPT * 256;
  const int b = blockIdx.x;
  const int tid = threadIdx.x;
  const float* X = xyz + (size_t)b * N * 3;
  float* O = ctrs + (size_t)b * S * 3;

  __shared__ float sval[256];
  __shared__ int   sidx[256];

  float dist[PPT];
#pragma unroll
  for (int j = 0; j < PPT; ++j) dist[j] = 1e10f;

  int last = 0;
  if (tid < 3) O[tid] = X[tid];  // first sample is always index 0

  for (int s = 1; s < S; ++s) {
    const float px = X[last*3+0], py = X[last*3+1], pz = X[last*3+2];
    float best = -1.0f; int bi = 0;
#pragma unroll 4
    for (int j = 0; j < PPT; ++j) {
      const int i = j * 256 + tid;
      const float dx = X[i*3+0]-px, dy = X[i*3+1]-py, dz = X[i*3+2]-pz;
      const float d = dx*dx + dy*dy + dz*dz;
      dist[j] = fminf(dist[j], d);
      if (dist[j] > best) { best = dist[j]; bi = i; }  // strict > keeps lowest i
    }
    sval[tid] = best; sidx[tid] = bi;
    __syncthreads();
    for (int off = 128; off > 0; off >>= 1) {
      if (tid < off) {
        const float v2 = sval[tid+off]; const int i2 = sidx[tid+off];
        if (v2 > sval[tid] || (v2 == sval[tid] && i2 < sidx[tid])) {
          sval[tid] = v2; sidx[tid] = i2;
        }
      }
      __syncthreads();
    }
    last = sidx[0];
    __syncthreads();
    if (tid < 3) O[s*3+tid] = X[last*3+tid];
  }
}

// ---------------------------------------------------------------------------
// Ball query: one wave32 per center. Ordered compaction via ballot+popcount
// reproduces "first nsample indices in index order, padded with first hit".
// ---------------------------------------------------------------------------
__global__ void ball_query_kernel(const float* __restrict__ xyz,
                                  const float* __restrict__ ctrs,
                                  int N, int S, int K, float r2,
                                  int* __restrict__ out) {
  const int gw = blockIdx.x * 8 + (threadIdx.x >> 5);
  if (gw >= BATCH * S) return;
  const int lane = threadIdx.x & 31;
  const int b = gw / S;
  const float* X = xyz + (size_t)b * N * 3;
  const float cx = ctrs[gw*3+0], cy = ctrs[gw*3+1], cz = ctrs[gw*3+2];
  int* O = out + (size_t)gw * K;

  int cnt = 0, first = -1;
  for (int base = 0; base < N; base += 32) {   // N is a multiple of 32
    const int i = base + lane;
    const float dx = X[i*3+0]-cx, dy = X[i*3+1]-cy, dz = X[i*3+2]-cz;
    const bool hit = (dx*dx + dy*dy + dz*dz) < r2;
    const unsigned m = __builtin_amdgcn_ballot_w32(hit);
    if (hit) {
      const int pos = cnt + __popc(m & ((1u << lane) - 1u));
      if (pos < K) O[pos] = i;
    }
    if (first < 0 && m != 0u) first = base + __ffs((int)m) - 1;
    cnt += __popc(m);
    if (cnt >= K) break;
  }
  for (int j = cnt + lane; j < K; j += 32) O[j] = first;  // pad with first hit
}

// ---------------------------------------------------------------------------
// SA1 MLP: per (b,s): X[64x3(rel)] -> relu(X@w0a+b0a)[64x64]
//          -> relu(.@w0b+b0b)[64x128] -> max over 64 -> feat1[b,s,128]
// block = 128 (4 waves, one 16-row M-tile each), grid = 16*256
// ---------------------------------------------------------------------------
__global__ void sa1_kernel(const float* __restrict__ pc,
                           const float* __restrict__ ctrs,
                           const int* __restrict__ bidx,
                           const float* __restrict__ w0a, const float* __restrict__ b0a,
                           const float* __restrict__ w0b, const float* __restrict__ b0b,
                           float* __restrict__ feat1) {
  const int g = blockIdx.x;
  const int b = g >> 8;                 // S = 256
  const int tid = threadIdx.x;
  const int wave = tid >> 5, lane = tid & 31;
  const int hi = lane >> 4, ln = lane & 15;
  const int m0 = wave * 16;

  __shared__ int   idxS[64];
  __shared__ float ctr[3];
  __shared__ float Xs[64*4];       // K padded 3->4
  __shared__ float h1[64*68];      // stride 68 (== 4 mod 64, conflict-free)
  __shared__ float pmax[8*132];

  __builtin_prefetch(w0b, 0, 3);
  if (tid < 64) idxS[tid] = bidx[g*64 + tid];
  if (tid < 3)  ctr[tid]  = ctrs[g*3 + tid];
  __syncthreads();
  for (int e = tid; e < 64*4; e += 128) {
    const int r = e >> 2, c = e & 3;
    Xs[e] = (c < 3) ? (pc[((size_t)b*NPTS + idxS[r])*3 + c] - ctr[c]) : 0.0f;
  }
  __syncthreads();

  // layer a: [64x4] @ [4(3)x64]
  for (int nt = 0; nt < 4; ++nt) {
    const int n0 = nt * 16;
    v8f acc = {};
    const int ka = 2*hi;
    v2f a, w;
    a.x = Xs[(m0+ln)*4 + ka];
    a.y = Xs[(m0+ln)*4 + ka + 1];
    w.x = (ka     < 3) ? w0a[ka*64     + n0+ln] : 0.0f;
    w.y = (ka + 1 < 3) ? w0a[(ka+1)*64 + n0+ln] : 0.0f;
    acc = wmma4(a, w, acc);
#pragma unroll
    for (int j = 0; j < 8; ++j)
      h1[(m0 + j + 8*hi)*68 + n0+ln] = fmaxf(acc[j] + b0a[n0+ln], 0.0f);
  }
  __syncthreads();

  // layer b: [64x64] @ [64x128], relu, row-max
  for (int nt = 0; nt < 8; ++nt) {
    const int n0 = nt * 16;
    v8f acc = {};
    for (int k0 = 0; k0 < 64; k0 += 4) {
      const int ka = k0 + 2*hi;
      v2f a, w;
      a.x = h1[(m0+ln)*68 + ka];
      a.y = h1[(m0+ln)*68 + ka + 1];
      w.x = w0b[ka*128     + n0+ln];
      w.y = w0b[(ka+1)*128 + n0+ln];
      acc = wmma4(a, w, acc);
    }
    const int col = n0 + ln;
    float pm = 0.0f;  // post-ReLU values are >= 0
#pragma unroll
    for (int j = 0; j < 8; ++j) pm = fmaxf(pm, fmaxf(acc[j] + b0b[col], 0.0f));
    pmax[(wave*2 + hi)*132 + col] = pm;
  }
  __syncthreads();
  if (tid < 128) {
    float v = pmax[tid];
#pragma unroll
    for (int r = 1; r < 8; ++r) v = fmaxf(v, pmax[r*132 + tid]);
    feat1[(size_t)g*128 + tid] = v;
  }
}

// ---------------------------------------------------------------------------
// SA2 MLP: per (b,s): X[128x131] = [rel xyz | gathered feat1]
//   -> relu(X@w1a+b1a)[128x128] -> relu(.@w1b+b1b)[128x256] -> max -> feat2
// block = 256 (8 waves), grid = 16*64. Xs/h1 in dynamic LDS (~135 KB).
// ---------------------------------------------------------------------------
__global__ void sa2_kernel(const float* __restrict__ xyz1,
                           const float* __restrict__ feat1,
                           const float* __restrict__ ctrs,
                           const int* __restrict__ bidx,
                           const float* __restrict__ w1a, const float* __restrict__ b1a,
                           const float* __restrict__ w1b, const float* __restrict__ b1b,
                           float* __restrict__ feat2) {
  const int g = blockIdx.x;
  const int b = g >> 6;                 // S = 64
  const int tid = threadIdx.x;
  const int wave = tid >> 5, lane = tid & 31;
  const int hi = lane >> 4, ln = lane & 15;
  const int m0 = wave * 16;

  extern __shared__ float smem[];
  float* Xs = smem;                 // [128][132], K padded 131->132
  float* h1 = smem + 128*132;       // [128][132], cols 0..127 used

  __shared__ int   idxS[128];
  __shared__ float ctr[3];
  __shared__ float pmax[16*260];

  __builtin_prefetch(w1b, 0, 3);
  if (tid < 128) idxS[tid] = bidx[g*128 + tid];
  if (tid < 3)   ctr[tid]  = ctrs[g*3 + tid];
  __syncthreads();
  for (int e = tid; e < 128*132; e += 256) {
    const int r = e / 132, c = e - r*132;
    const int pi = idxS[r];
    float v;
    if (c < 3)        v = xyz1[((size_t)b*256 + pi)*3 + c] - ctr[c];
    else if (c < 131) v = feat1[((size_t)b*256 + pi)*128 + (c-3)];
    else              v = 0.0f;
    Xs[e] = v;
  }
  __syncthreads();

  // layer a: [128x132(131)] @ [131x128]
  for (int nt = 0; nt < 8; ++nt) {
    const int n0 = nt * 16;
    v8f acc = {};
    for (int k0 = 0; k0 < 132; k0 += 4) {
      const int ka = k0 + 2*hi;
      v2f a, w;
      a.x = Xs[(m0+ln)*132 + ka];
      a.y = Xs[(m0+ln)*132 + ka + 1];
      w.x = (ka     < 131) ? w1a[ka*128     + n0+ln] : 0.0f;
      w.y = (ka + 1 < 131) ? w1a[(ka+1)*128 + n0+ln] : 0.0f;
      acc = wmma4(a, w, acc);
    }
#pragma unroll
    for (int j = 0; j < 8; ++j)
      h1[(m0 + j + 8*hi)*132 + n0+ln] = fmaxf(acc[j] + b1a[n0+ln], 0.0f);
  }
  __syncthreads();

  // layer b: [128x128] @ [128x256], relu, row-max
  for (int nt = 0; nt < 16; ++nt) {
    const int n0 = nt * 16;
    v8f acc = {};
    for (int k0 = 0; k0 < 128; k0 += 4) {
      const int ka = k0 + 2*hi;
      v2f a, w;
      a.x = h1[(m0+ln)*132 + ka];
      a.y = h1[(m0+ln)*132 + ka + 1];
      w.x = w1b[ka*256     + n0+ln];
      w.y = w1b[(ka+1)*256 + n0+ln];
      acc = wmma4(a, w, acc);
    }
    const int col = n0 + ln;
    float pm = 0.0f;
#pragma unroll
    for (int j = 0; j < 8; ++j) pm = fmaxf(pm, fmaxf(acc[j] + b1b[col], 0.0f));
    pmax[(wave*2 + hi)*260 + col] = pm;
  }
  __syncthreads();
  {
    const int col = tid;  // 256 threads == 256 cols
    float v = pmax[col];
#pragma unroll
    for (int r = 1; r < 16; ++r) v = fmaxf(v, pmax[r*260 + col]);
    feat2[(size_t)g*256 + col] = v;
  }
}

// ---------------------------------------------------------------------------
// SA-all: per batch: X[64x259] = [xyz2 | feat2] (no centering, no gather)
//   -> relu(X@w2a+b2a)[64x256] -> relu(.@w2b+b2b)[64x512] -> max -> feat3[b,512]
// block = 128 (4 waves), grid = 16
// ---------------------------------------------------------------------------
__global__ void sa3_kernel(const float* __restrict__ xyz2,
                           const float* __restrict__ feat2,
                           const float* __restrict__ w2a, const float* __restrict__ b2a,
                           const float* __restrict__ w2b, const float* __restrict__ b2b,
                           float* __restrict__ feat3) {
  const int b = blockIdx.x;
  const int tid = threadIdx.x;
  const int wave = tid >> 5, lane = tid & 31;
  const int hi = lane >> 4, ln = lane & 15;
  const int m0 = wave * 16;

  extern __shared__ float smem[];
  float* Xs = smem;               // [64][260], K padded 259->260
  float* h1 = smem + 64*260;      // [64][260], cols 0..255 used

  __shared__ float pmax[8*516];

  __builtin_prefetch(w2b, 0, 3);
  for (int e = tid; e < 64*260; e += 128) {
    const int r = e / 260, c = e - r*260;
    float v;
    if (c < 3)        v = xyz2[((size_t)b*64 + r)*3 + c];
    else if (c < 259) v = feat2[((size_t)b*64 + r)*256 + (c-3)];
    else              v = 0.0f;
    Xs[e] = v;
  }
  __syncthreads();

  // layer a: [64x260(259)] @ [259x256]
  for (int nt = 0; nt < 16; ++nt) {
    const int n0 = nt * 16;
    v8f acc = {};
    for (int k0 = 0; k0 < 260; k0 += 4) {
      const int ka = k0 + 2*hi;
      v2f a, w;
      a.x = Xs[(m0+ln)*260 + ka];
      a.y = Xs[(m0+ln)*260 + ka + 1];
      w.x = (ka     < 259) ? w2a[ka*256     + n0+ln] : 0.0f;
      w.y = (ka + 1 < 259) ? w2a[(ka+1)*256 + n0+ln] : 0.0f;
      acc = wmma4(a, w, acc);
    }
#pragma unroll
    for (int j = 0; j < 8; ++j)
      h1[(m0 + j + 8*hi)*260 + n0+ln] = fmaxf(acc[j] + b2a[n0+ln], 0.0f);
  }
  __syncthreads();

  // layer b: [64x256] @ [256x512], relu, row-max
  for (int nt = 0; nt < 32; ++nt) {
    const int n0 = nt * 16;
    v8f acc = {};
    for (int k0 = 0; k0 < 256; k0 += 4) {
      const int ka = k0 + 2*hi;
      v2f a, w;
      a.x = h1[(m0+ln)*260 + ka];
      a.y = h1[(m0+ln)*260 + ka + 1];
      w.x = w2b[ka*512     + n0+ln];
      w.y = w2b[(ka+1)*512 + n0+ln];
      acc = wmma4(a, w, acc);
    }
    const int col = n0 + ln;
    float pm = 0.0f;
#pragma unroll
    for (int j = 0; j < 8; ++j) pm = fmaxf(pm, fmaxf(acc[j] + b2b[col], 0.0f));
    pmax[(wave*2 + hi)*516 + col] = pm;
  }
  __syncthreads();
  for (int col = tid; col < 512; col += 128) {
    float v = pmax[col];
#pragma unroll
    for (int r = 1; r < 8; ++r) v = fmaxf(v, pmax[r*516 + col]);
    feat3[(size_t)b*512 + col] = v;
  }
}

// ---------------------------------------------------------------------------
// FC head: [16x512] -> relu fc0 -> [16x1024] -> relu fc1 -> out [16x1024]
// grid = 1, block = 256 (8 waves share the single 16-row M-tile, split N)
// ---------------------------------------------------------------------------
__global__ void fc_kernel(const float* __restrict__ feat3,
                          const float* __restrict__ fc0w, const float* __restrict__ fc0b,
                          const float* __restrict__ fc1w, const float* __restrict__ fc1b,
                          float* __restrict__ out) {
  const int tid = threadIdx.x;
  const int wave = tid >> 5, lane = tid & 31;
  const int hi = lane >> 4, ln = lane & 15;

  extern __shared__ float smem[];
  float* h0 = smem;               // [16][1028]
  __shared__ float xS[16*516];    // [16][516], cols 0..511 used

  __builtin_prefetch(fc1w, 0, 3);
  for (int e = tid; e < 16*516; e += 256) {
    const int r = e / 516, c = e - r*516;
    xS[e] = (c < 512) ? feat3[r*512 + c] : 0.0f;
  }
  __syncthreads();

  for (int nt = wave; nt < 64; nt += 8) {    // fc0: N = 1024
    const int n0 = nt * 16;
    v8f acc = {};
    for (int k0 = 0; k0 < 512; k0 += 4) {
      const int ka = k0 + 2*hi;
      v2f a, w;
      a.x = xS[ln*516 + ka];
      a.y = xS[ln*516 + ka + 1];
      w.x = fc0w[ka*1024     + n0+ln];
      w.y = fc0w[(ka+1)*1024 + n0+ln];
      acc = wmma4(a, w, acc);
    }
#pragma unroll
    for (int j = 0; j < 8; ++j)
      h0[(j + 8*hi)*1028 + n0+ln] = fmaxf(acc[j] + fc0b[n0+ln], 0.0f);
  }
  __syncthreads();

  for (int nt = wave; nt < 64; nt += 8) {    // fc1: N = 1024
    const int n0 = nt * 16;
    v8f acc = {};
    for (int k0 = 0; k0 < 1024; k0 += 4) {
      const int ka = k0 + 2*hi;
      v2f a, w;
      a.x = h0[ln*1028 + ka];
      a.y = h0[ln*1028 + ka + 1];
      w.x = fc1w[ka*1024     + n0+ln];
      w.y = fc1w[(ka+1)*1024 + n0+ln];
      acc = wmma4(a, w, acc);
    }
#pragma unroll
    for (int j = 0; j < 8; ++j)
      out[(size_t)(j + 8*hi)*1024 + n0+ln] = fmaxf(acc[j] + fc1b[n0+ln], 0.0f);
  }
}

// ---------------------------------------------------------------------------
extern "C" void kernel_launch(void* const* d_in, const int* in_sizes, int n_in,
                              void* d_out, int out_size, void* d_ws, size_t ws_size,
                              hipStream_t stream) {
  const float* pc   = (const float*)d_in[0];
  const float* w0a  = (const float*)d_in[1];
  const float* b0a  = (const float*)d_in[2];
  const float* w0b  = (const float*)d_in[3];
  const float* b0b  = (const float*)d_in[4];
  const float* w1a  = (const float*)d_in[5];
  const float* b1a  = (const float*)d_in[6];
  const float* w1b  = (const float*)d_in[7];
  const float* b1b  = (const float*)d_in[8];
  const float* w2a  = (const float*)d_in[9];
  const float* b2a  = (const float*)d_in[10];
  const float* w2b  = (const float*)d_in[11];
  const float* b2b  = (const float*)d_in[12];
  const float* fc0w = (const float*)d_in[13];
  const float* fc0b = (const float*)d_in[14];
  const float* fc1w = (const float*)d_in[15];
  const float* fc1b = (const float*)d_in[16];
  float* outp = (float*)d_out;

  // workspace layout (floats), ~4.8 MB total
  float* ws    = (float*)d_ws;
  float* nx1   = ws;                                   // 16*256*3
  int*   idx1  = (int*)(nx1 + 16*256*3);               // 16*256*64
  float* feat1 = (float*)(idx1 + 16*256*64);           // 16*256*128
  float* nx2   = feat1 + 16*256*128;                   // 16*64*3
  int*   idx2  = (int*)(nx2 + 16*64*3);                // 16*64*128
  float* feat2 = (float*)(idx2 + 16*64*128);           // 16*64*256
  float* feat3 = feat2 + 16*64*256;                    // 16*512

  // SA1: FPS 8192->256, ball r=0.02 K=64, MLP 3->64->128, pool
  fps_kernel<32><<<16, 256, 0, stream>>>(pc, 256, nx1);
  ball_query_kernel<<<(16*256 + 7)/8, 256, 0, stream>>>(
      pc, nx1, 8192, 256, 64, 0.02f*0.02f, idx1);
  sa1_kernel<<<16*256, 128, 0, stream>>>(pc, nx1, idx1, w0a, b0a, w0b, b0b, feat1);

  // SA2: FPS 256->64, ball r=0.04 K=128, MLP 131->128->256, pool
  fps_kernel<1><<<16, 256, 0, stream>>>(nx1, 64, nx2);
  ball_query_kernel<<<(16*64 + 7)/8, 256, 0, stream>>>(
      nx1, nx2, 256, 64, 128, 0.04f*0.04f, idx2);
  sa2_kernel<<<16*64, 256, (size_t)(2*128*132)*sizeof(float), stream>>>(
      nx1, feat1, nx2, idx2, w1a, b1a, w1b, b1b, feat2);

  // SA-all: MLP 259->256->512, pool over 64
  sa3_kernel<<<16, 128, (size_t)(2*64*260)*sizeof(float), stream>>>(
      nx2, feat2, w2a, b2a, w2b, b2b, feat3);

  // FC head
  fc_kernel<<<1, 256, (size_t)(16*1028)*sizeof(float), stream>>>(
      feat3, fc0w, fc0b, fc1w, fc1b, outp);
}